// Downsample2d_22943715295932
// MI455X (gfx1250) — compile-verified
//
#include <hip/hip_runtime.h>
#include <cstdint>

// Depthwise 4x4 blur, stride 2, reflect pad=1.
// [8,256,256,256] f32 -> [8,256,128,128] f32.
// Memory-bound (~0.67 GB @ 23.3 TB/s ~= 29us floor). Strategy: async
// global->LDS tile staging (CDNA5 GLOBAL_LOAD_ASYNC_TO_LDS_*, ASYNCcnt),
// exactly-once input reads, 128-bit coalesced loads, NT stores.
//
// Round-2 fix: the LDS destination address fed to the async-load asm is now
// derived from &tile[...] (ptrtoint -> low 32 bits == AS(3) offset). This
// makes `tile` escape, so the "memory" clobber actually aliases it and the
// compiler can no longer fold the LDS-consuming FMA tree to undef.

#define H   256
#define W   256
#define OH  128
#define OW  128
#define TOUT 16                 // output rows per tile
#define INR  34                 // input rows per tile: 2*16 + 2
#define LSTRIDE 264             // floats per LDS row (data at col offset +4 -> 16B aligned)

__device__ __forceinline__ int reflect256(int i) {
    i = (i < 0) ? -i : i;            // reflect(-1) = 1
    i = (i > 255) ? (510 - i) : i;   // reflect(256) = 254
    return i;
}

__global__ __launch_bounds__(256) void downsample2d_blur_kernel(
    const float* __restrict__ x,     // [planes, 256, 256]
    const float* __restrict__ kern,  // [4,4]
    float* __restrict__ out)         // [planes, 128, 128]
{
    __shared__ float tile[INR * LSTRIDE];   // ~35 KB

    const int tid   = threadIdx.x;
    const int plane = blockIdx.x >> 3;
    const int r0out = (blockIdx.x & 7) * TOUT;   // first output row of tile
    const int rin0  = r0out * 2 - 1;             // first (virtual) input row

    const float* px = x + (size_t)plane * (H * W);

    // Low 32 bits of the generic address of LDS == byte offset in the LDS
    // aperture, which is what the async-load VDST VGPR expects. Taking the
    // address here also forces `tile` to escape (see header comment).
    const unsigned tile_base = (unsigned)(uintptr_t)(&tile[0]);

    // ---- async main-body loads: 34 rows x 64 float4 columns ----------------
    // LDS addr = tile_base + (r*LSTRIDE + 4 + 4*c4)*4 bytes -> 16B aligned.
    // Global addr = plane_base + (gr*256 + 4*c4)*4 -> 16B aligned.
    #pragma unroll
    for (int i = 0; i < 9; ++i) {
        int idx = i * 256 + tid;
        if (idx < INR * 64) {
            int r  = idx >> 6;
            int c4 = idx & 63;
            int gr = reflect256(rin0 + r);
            const float* gp = px + gr * W + c4 * 4;
            unsigned lds_addr = tile_base + (unsigned)((r * LSTRIDE + 4 + c4 * 4) * 4);
            asm volatile("global_load_async_to_lds_b128 %0, %1, off"
                         :: "v"(lds_addr), "v"(gp) : "memory");
        }
    }
    // ---- halo columns: col -1 -> reflect col 1 ; col 256 -> reflect col 254
    if (tid < INR * 2) {
        int r    = tid >> 1;
        int side = tid & 1;
        int gr   = reflect256(rin0 + r);
        int gc   = side ? 254 : 1;            // reflected global column
        int lc   = side ? (4 + 256) : 3;      // LDS float index within row
        const float* gp = px + gr * W + gc;
        unsigned lds_addr = tile_base + (unsigned)((r * LSTRIDE + lc) * 4);
        asm volatile("global_load_async_to_lds_b32 %0, %1, off"
                     :: "v"(lds_addr), "v"(gp) : "memory");
    }

    // Wait for this wave's async transfers, then sync the workgroup so every
    // wave sees all LDS tiles.
    asm volatile("s_wait_asynccnt 0" ::: "memory");
    __syncthreads();

    // ---- 4x4 taps (uniform; compiler keeps these scalar) -------------------
    float k[16];
    #pragma unroll
    for (int j = 0; j < 16; ++j) k[j] = kern[j];

    // ---- compute: 256 threads x 8 outputs = 16 x 128 tile ------------------
    const int c    = tid & 127;   // output column
    const int rg   = tid >> 7;    // 0 or 1 -> rows [0..7] or [8..15]
    float* pout = out + (size_t)plane * (OH * OW);

    #pragma unroll
    for (int q = 0; q < 8; ++q) {
        int rr = rg * 8 + q;                                  // tile-local out row
        // out(rr, c) uses LDS rows 2rr..2rr+3, float cols (2c+3)..(2c+6)
        const float* base = &tile[(2 * rr) * LSTRIDE + 2 * c + 3];
        float acc = 0.f;
        #pragma unroll
        for (int i = 0; i < 4; ++i) {
            #pragma unroll
            for (int j = 0; j < 4; ++j)
                acc = fmaf(k[i * 4 + j], base[i * LSTRIDE + j], acc);
        }
        // write-once output: non-temporal store, keep L2 for the input stream
        __builtin_nontemporal_store(acc, &pout[(size_t)(r0out + rr) * OW + c]);
    }
}

extern "C" void kernel_launch(void* const* d_in, const int* in_sizes, int n_in,
                              void* d_out, int out_size, void* d_ws, size_t ws_size,
                              hipStream_t stream) {
    const float* x    = (const float*)d_in[0];
    const float* kern = (const float*)d_in[1];
    float* out        = (float*)d_out;

    int planes = in_sizes[0] / (H * W);      // B*C = 2048
    dim3 grid(planes * 8);                   // 8 row-tiles of 16 output rows
    downsample2d_blur_kernel<<<grid, 256, 0, stream>>>(x, kern, out);
}